// PhysicsLoss_56865366999461
// MI455X (gfx1250) — compile-verified
//
#include <hip/hip_runtime.h>
#include <hip/hip_bf16.h>

// ---------------------------------------------------------------------------
// PhysicsLoss for MI455X (gfx1250, wave32).
// Bandwidth-bound: ~117 MB HBM traffic -> ~5us floor @ 23.3 TB/s.
// 32x32 tiles staged in LDS. Interior tiles use GLOBAL_LOAD_ASYNC_TO_LDS
// (raw data; means cancel analytically without padding). Boundary tiles use
// a scalar pre-normalized, zero-padded fill. Per-wave reduction via
// v_wmma_f32_16x16x4_f32 (B = ones => row sums), deterministic 2-pass reduce.
// ---------------------------------------------------------------------------

typedef __attribute__((ext_vector_type(2))) float v2f;
typedef __attribute__((ext_vector_type(8))) float v8f;

#if __has_builtin(__builtin_amdgcn_global_load_async_to_lds_b32)
#define HAVE_ASYNC_LDS 1
typedef __attribute__((address_space(1))) int g_int_t;   // global (cuda_device)
typedef __attribute__((address_space(3))) int l_int_t;   // LDS
#else
#define HAVE_ASYNC_LDS 0
#endif

#define HH 512
#define WW 512
#define NB 16
#define TILE 32
#define PLANE (HH * WW)          // 262144
#define BSTRIDE (4 * PLANE)      // per-batch stride
#define NBLOCKS (16 * 16 * NB)   // 4096 tile blocks

__global__ __launch_bounds__(256) void physics_tile_kernel(
    const float* __restrict__ f_now,
    const float* __restrict__ f_next,
    const float* __restrict__ dtp,
    float* __restrict__ block_sums)
{
    // Interior tiles: g holds RAW f_next. Boundary tiles: g holds
    // f_next*std+mean with zero "SAME" padding outside the image.
    __shared__ float g[4][TILE + 2][TILE + 4];   // row stride 36 (bank spread)
    __shared__ float wsum[16];

    const float STDS[4]  = {0.1f, 0.1f, 0.2f, 1.0f};
    const float MEANS[4] = {0.0f, 0.0f, 0.5f, 0.0f};
    const float inv0 = 1.0f / (0.1f + 1e-8f);
    const float inv1 = 1.0f / (0.1f + 1e-8f);
    const float inv2 = 1.0f / (0.2f + 1e-8f);
    const float inv3 = 1.0f / (1.0f + 1e-8f);
    const float PR = 0.71f;
    const float PRDA = 71.0f;      // PR/DA
    const float RAPR = 71000.0f;   // RA*PR
    const float HA2PR = 71.0f;     // HA^2*PR
    const float C73 = 7.0f / 3.0f; // 1 + 4*RD/3
    const float Qc = 1.0f;

    const int tid = threadIdx.x;
    const int tileX0 = blockIdx.x * TILE;
    const int tileY0 = blockIdx.y * TILE;
    const int b = blockIdx.z;
    const long bbase = (long)b * BSTRIDE;

    const bool interior = (tileX0 > 0) && (tileY0 > 0) &&
                          (tileX0 + TILE < WW) && (tileY0 + TILE < HH);
#if HAVE_ASYNC_LDS
    const bool rawLds = interior;
#else
    const bool rawLds = false;
#endif

    // ---- stage f_next halo tile into LDS ----
#if HAVE_ASYNC_LDS
    if (rawLds) {
        // Interior: DMA raw values straight into LDS (ASYNCcnt path).
        for (int i = tid; i < 4 * 34 * 34; i += 256) {
            const int c  = i / (34 * 34);
            const int r  = i - c * (34 * 34);
            const int ly = r / 34;
            const int lx = r - ly * 34;
            const float* src = f_next + bbase + (long)c * PLANE +
                               (long)(tileY0 + ly - 1) * WW + (tileX0 + lx - 1);
            __builtin_amdgcn_global_load_async_to_lds_b32(
                (g_int_t*)src, (l_int_t*)&g[c][ly][lx], 0, 0);
        }
#if __has_builtin(__builtin_amdgcn_s_wait_asynccnt)
        __builtin_amdgcn_s_wait_asynccnt(0);
#else
        asm volatile("s_wait_asynccnt 0" ::: "memory");
#endif
    } else
#endif
    {
        // Boundary: scalar fill, pre-normalized, zero-padded.
        for (int i = tid; i < 4 * 34 * 34; i += 256) {
            const int c  = i / (34 * 34);
            const int r  = i - c * (34 * 34);
            const int ly = r / 34;
            const int lx = r - ly * 34;
            const int gy = tileY0 + ly - 1;
            const int gx = tileX0 + lx - 1;
            float v = 0.0f;
            if (gy >= 0 && gy < HH && gx >= 0 && gx < WW) {
                v = f_next[bbase + (long)c * PLANE + (long)gy * WW + gx] * STDS[c] + MEANS[c];
            }
            g[c][ly][lx] = v;
        }
    }
    __syncthreads();

    const float rdt = 1.0f / dtp[0];

    // Block-uniform constants folding the raw-vs-scaled LDS representation.
    // rawLds: LDS holds raw x  => effective value = x*std + mean; means cancel
    //         in all derivatives (no padding), centers fold to raw*std*inv.
    const float kc0 = rawLds ? 0.1f : 1.0f;
    const float kc1 = rawLds ? 0.1f : 1.0f;
    const float kc2 = rawLds ? 0.2f : 1.0f;
    const float O2v = rawLds ? 0.0f : 0.5f;   // T-channel center offset
    const float C0 = inv0 * kc0, C1 = inv1 * kc1, C2 = inv2 * kc2;
    const float DXs0 = C0 * 50.0f, DXs1 = C1 * 50.0f, DXs2 = C2 * 50.0f;
    const float DXs3 = inv3 * 50.0f;                   // P: std=1 both paths
    const float LPs0 = C0 * 1.0e4f, LPs1 = C1 * 1.0e4f, LPs2 = C2 * 1.0e4f;
    const float FT0 = 0.1f * inv0 * rdt;   // f_now multipliers (path-invariant)
    const float FT1 = 0.1f * inv1 * rdt;
    const float FT2 = 0.2f * inv2 * rdt;

    float acc = 0.0f;

    // each thread: 4 pixels of the 32x32 tile
    #pragma unroll
    for (int k = 0; k < 4; ++k) {
        const int p  = tid + k * 256;
        const int py = p >> 5;
        const int px = p & 31;
        const int ly = py + 1;
        const int lx = px + 1;

        const float u_c = g[0][ly][lx],  u_n = g[0][ly-1][lx], u_s = g[0][ly+1][lx];
        const float u_w = g[0][ly][lx-1], u_e = g[0][ly][lx+1];
        const float v_c = g[1][ly][lx],  v_n = g[1][ly-1][lx], v_s = g[1][ly+1][lx];
        const float v_w = g[1][ly][lx-1], v_e = g[1][ly][lx+1];
        const float t_c = g[2][ly][lx],  t_n = g[2][ly-1][lx], t_s = g[2][ly+1][lx];
        const float t_w = g[2][ly][lx-1], t_e = g[2][ly][lx+1];
        const float p_n = g[3][ly-1][lx], p_s = g[3][ly+1][lx];
        const float p_w = g[3][ly][lx-1], p_e = g[3][ly][lx+1];

        const long gidx = bbase + (long)(tileY0 + py) * WW + (tileX0 + px);
        const float f0U = f_now[gidx];
        const float f0V = f_now[gidx + PLANE];
        const float f0T = f_now[gidx + 2 * PLANE];

        // dfdt = (f1 - f0)*inv/dt with both LDS representations unified
        const float dUdt = u_c * C0 * rdt - f0U * FT0;
        const float dVdt = v_c * C1 * rdt - f0V * FT1;
        const float dTdt = (t_c - O2v) * C2 * rdt - f0T * FT2;

        const float dUdx = (u_e - u_w) * DXs0, dUdy = (u_s - u_n) * DXs0;
        const float dVdx = (v_e - v_w) * DXs1, dVdy = (v_s - v_n) * DXs1;
        const float dTdx = (t_e - t_w) * DXs2, dTdy = (t_s - t_n) * DXs2;
        const float dPdx = (p_e - p_w) * DXs3, dPdy = (p_s - p_n) * DXs3;

        const float lapU = (u_n + u_s + u_e + u_w - 4.0f * u_c) * LPs0;
        const float lapV = (v_n + v_s + v_e + v_w - 4.0f * v_c) * LPs1;
        const float lapT = (t_n + t_s + t_e + t_w - 4.0f * t_c) * LPs2;

        const float U1 = u_c * C0;
        const float V1 = v_c * C1;
        const float T1 = (t_c - O2v) * C2;

        const float r1 = dUdx + dVdy;
        const float r2 = dUdt + U1 * dUdx + V1 * dUdy + dPdx - PR * lapU - PRDA * U1;
        const float r3 = dVdt + U1 * dVdx + V1 * dVdy + dPdy - PR * lapV
                         - RAPR * T1 + HA2PR * V1 - PRDA * V1;
        const float r4 = dTdt + U1 * dTdx + V1 * dTdy - C73 * lapT - Qc * T1;

        acc += r1 * r1 + r2 * r2 + r3 * r3 + r4 * r4;
    }

    // ---- per-wave reduction via WMMA: D = A x Ones => D rows = row-sums of A.
    // K-slots 1/3 zeroed, so K-layout details cannot affect the total.
    v2f a;  a.x = acc;  a.y = 0.0f;
    v2f bo; bo.x = 1.0f; bo.y = 1.0f;
    v8f cz = {};
    v8f d = __builtin_amdgcn_wmma_f32_16x16x4_f32(
        false, a, false, bo, (short)0, cz, false, false);
    float s = d[0] + d[1] + d[2] + d[3] + d[4] + d[5] + d[6] + d[7];
    // lanes 0..15 hold sum(rows 0..7); lanes 16..31 hold sum(rows 8..15)

    const int wid = tid >> 5;
    const int ln  = tid & 31;
    if (ln == 0)  wsum[2 * wid + 0] = s;
    if (ln == 16) wsum[2 * wid + 1] = s;
    __syncthreads();

    if (tid == 0) {
        float t = 0.0f;
        #pragma unroll
        for (int i = 0; i < 16; ++i) t += wsum[i];
        block_sums[(blockIdx.z * 16 + blockIdx.y) * 16 + blockIdx.x] = t;
    }
}

__global__ __launch_bounds__(256) void physics_reduce_kernel(
    const float* __restrict__ block_sums, float* __restrict__ out)
{
    __shared__ float sm[256];
    float s = 0.0f;
    for (int i = threadIdx.x; i < NBLOCKS; i += 256) s += block_sums[i];
    sm[threadIdx.x] = s;
    __syncthreads();
    for (int off = 128; off > 0; off >>= 1) {
        if (threadIdx.x < off) sm[threadIdx.x] += sm[threadIdx.x + off];
        __syncthreads();
    }
    if (threadIdx.x == 0) {
        // mean over B*C*H*W = 16*4*512*512 = 16777216 elements
        out[0] = sm[0] * (1.0f / 16777216.0f);
    }
}

extern "C" void kernel_launch(void* const* d_in, const int* in_sizes, int n_in,
                              void* d_out, int out_size, void* d_ws, size_t ws_size,
                              hipStream_t stream) {
    (void)in_sizes; (void)n_in; (void)out_size; (void)ws_size;
    const float* f_now  = (const float*)d_in[0];
    const float* f_next = (const float*)d_in[1];
    const float* dtp    = (const float*)d_in[2];
    float* out = (float*)d_out;
    float* ws  = (float*)d_ws;   // 4096 floats of block partial sums

    dim3 grid(WW / TILE, HH / TILE, NB);   // 16 x 16 x 16
    physics_tile_kernel<<<grid, 256, 0, stream>>>(f_now, f_next, dtp, ws);
    physics_reduce_kernel<<<1, 256, 0, stream>>>(ws, out);
}